// HomogeneousTiles_64029372448827
// MI455X (gfx1250) — compile-verified
//
#include <hip/hip_runtime.h>
#include <stdint.h>

// Problem constants (match reference)
#define HH 4001
#define WW 4001
#define TT 64             // tile size
#define ST 8              // stride
#define NL 493            // (4001-64)/8 + 1 tile corners per dim
#define NTILE (NL * NL)   // 243049
#define RS_PITCH 512      // padded pitch for rowsum arrays (>= NL + 3 for edge tiles)
#define NROWS_PAD 4032    // rows 4001..4031 zero-filled (band matmul reads up to 4024)
#define PSEL 64           // patches selected
#define SIMG 512          // output image side

typedef float v2f __attribute__((ext_vector_type(2)));
typedef float v8f __attribute__((ext_vector_type(8)));

// ---------------------------------------------------------------------------
// Kernel 1: gray conversion + 64-wide / stride-8 row window sums (and squares)
// One block per (padded) image row; gray row staged in LDS.
// Rows >= HH are explicitly zeroed so the WMMA K-padding multiplies 0*0, never
// 0*NaN (first call runs on unpoisoned workspace memory).
// ---------------------------------------------------------------------------
__global__ __launch_bounds__(256) void k_rowsum(const float* __restrict__ img,
                                                float* __restrict__ rs1,
                                                float* __restrict__ rs2) {
  const int h = blockIdx.x;
  if (h >= HH) {  // zero-fill padding rows
    for (int jw = threadIdx.x; jw < NL; jw += blockDim.x) {
      rs1[(size_t)h * RS_PITCH + jw] = 0.0f;
      rs2[(size_t)h * RS_PITCH + jw] = 0.0f;
    }
    return;
  }
  __shared__ float g[WW];
  const float* rp = img + (size_t)0 * HH * WW + (size_t)h * WW;
  const float* gp = img + (size_t)1 * HH * WW + (size_t)h * WW;
  const float* bp = img + (size_t)2 * HH * WW + (size_t)h * WW;
  for (int w = threadIdx.x; w < WW; w += blockDim.x)
    g[w] = 0.2989f * rp[w] + 0.587f * gp[w] + 0.114f * bp[w];
  __syncthreads();
  for (int jw = threadIdx.x; jw < NL; jw += blockDim.x) {
    float s1 = 0.f, s2 = 0.f;
    const int base = jw * ST;
#pragma unroll 8
    for (int x = 0; x < TT; ++x) {
      float v = g[base + x];
      s1 += v;
      s2 = fmaf(v, v, s2);
    }
    rs1[(size_t)h * RS_PITCH + jw] = s1;
    rs2[(size_t)h * RS_PITCH + jw] = s2;
  }
}

// ---------------------------------------------------------------------------
// Kernel 2: column reduction as banded-matrix WMMA + std -> sortable keys
// One wave32 per 16x16 tile of (ih, jw):
//   out[ih,jw] = sum_{r=1..64} rowsum[8*ih+r][jw]  ==  C = A x B,
//   A[m,k] = 1 iff 8m <= k < 8m+64 (K = 184 tiled by 4), B[k,n] = rowsum row.
// f32 16x16x4 lane layouts (ISA 7.12.2): A/B VGPR0 holds K={0|2}, VGPR1 K={1|3}
// split across half-waves; the half-wave K offset (2*hi) is folded into the
// base pointer so every load has a compile-time immediate offset.
// Band edges are multiples of 8 and kx is even => a.x == a.y, one range test.
// ---------------------------------------------------------------------------
__global__ __launch_bounds__(32) void k_std_wmma(const float* __restrict__ rs1,
                                                 const float* __restrict__ rs2,
                                                 unsigned long long* __restrict__ keys) {
  const int lane = (int)threadIdx.x;
  const int m = lane & 15;
  const int hi = lane >> 4;  // 0 or 1
  const int ihb = blockIdx.y * 16;
  const int jwb = blockIdx.x * 16;
  const int jw = jwb + m;

  // Per-lane base pointers: row (8*ihb + 1 + 2*hi), column jw. All subsequent
  // rows are compile-time multiples of RS_PITCH (fully unrolled loop).
  const size_t base = (size_t)(8 * ihb + 1 + 2 * hi) * RS_PITCH + (size_t)jw;
  const float* p1 = rs1 + base;
  const float* p2 = rs2 + base;
  const int s = 8 * m - 2 * hi;  // band start in "4c" units

  v8f c1 = {};
  v8f c2 = {};

#pragma unroll
  for (int c = 0; c < 46; ++c) {  // K = 184 = 46 * 4
    const float av = ((unsigned)(4 * c - s) < 64u) ? 1.0f : 0.0f;
    v2f a;
    a.x = av;
    a.y = av;
    v2f b1, b2;
    b1.x = p1[(size_t)(4 * c + 0) * RS_PITCH];
    b1.y = p1[(size_t)(4 * c + 2) * RS_PITCH];  // overwritten below; K=+1 within chunk lives 1 row up
    b2.x = p2[(size_t)(4 * c + 0) * RS_PITCH];
    b2.y = p2[(size_t)(4 * c + 2) * RS_PITCH];
    // NOTE: VGPR1 of A/B holds K = {1 | 3}; with the 2*hi fold the .y element
    // is row base + 4c + 2 for BOTH half-waves? No: .y k = kx+1 -> row +1.
    b1.y = p1[(size_t)(4 * c + 1) * RS_PITCH];
    b2.y = p2[(size_t)(4 * c + 1) * RS_PITCH];
#if __has_builtin(__builtin_amdgcn_wmma_f32_16x16x4_f32)
    c1 = __builtin_amdgcn_wmma_f32_16x16x4_f32(false, a, false, b1, (short)0, c1,
                                               false, false);
    c2 = __builtin_amdgcn_wmma_f32_16x16x4_f32(false, a, false, b2, (short)0, c2,
                                               false, false);
#else
    for (int r = 0; r < 8; ++r) {
      c1[r] += a.x * b1.x + a.y * b1.y;
      c2[r] += a.x * b2.x + a.y * b2.y;
    }
#endif
  }

  const float nn = (float)(TT * TT);
#pragma unroll
  for (int r = 0; r < 8; ++r) {
    const int ih = ihb + r + 8 * hi;  // C elem r: M = r + 8*hi, N = lane&15
    if (ih < NL && jw < NL) {
      const float s1 = c1[r];
      const float s2 = c2[r];
      const float sd = sqrtf((s2 - s1 * s1 / nn) / nn);  // NaN if var<0 -> sorts last
      const unsigned int fb = __float_as_uint(sd);       // monotone for sd >= 0
      const unsigned int idx = (unsigned int)(ih * NL + jw);
      keys[idx] = ((unsigned long long)fb << 32) | (unsigned long long)idx;
    }
  }
}

// ---------------------------------------------------------------------------
// Kernel 3: stable 64-smallest selection (keys unique -> ascending min passes)
// ---------------------------------------------------------------------------
__device__ __forceinline__ unsigned long long ullmin2(unsigned long long a,
                                                      unsigned long long b) {
  return a < b ? a : b;
}

__global__ __launch_bounds__(1024) void k_select(const unsigned long long* __restrict__ keys,
                                                 unsigned int* __restrict__ sel) {
  __shared__ unsigned long long red[1024];
  const int tid = threadIdx.x;
  unsigned long long prev = 0ull;
  for (int p = 0; p < PSEL; ++p) {
    unsigned long long lm = ~0ull;
    for (int i = tid; i < NTILE; i += 1024) {
      const unsigned long long k = keys[i];
      if ((p == 0 || k > prev) && k < lm) lm = k;
    }
    red[tid] = lm;
    __syncthreads();
    for (int sstep = 512; sstep > 0; sstep >>= 1) {
      if (tid < sstep) red[tid] = ullmin2(red[tid], red[tid + sstep]);
      __syncthreads();
    }
    prev = red[0];
    if (tid == 0) sel[p] = (unsigned int)(prev & 0xFFFFFFFFull);
    __syncthreads();
  }
}

// ---------------------------------------------------------------------------
// Kernel 4: gather selected patches into the 3x512x512 fold layout
// ---------------------------------------------------------------------------
__global__ __launch_bounds__(256) void k_gather(const float* __restrict__ img,
                                                const unsigned int* __restrict__ sel,
                                                float* __restrict__ out) {
  const int tid = blockIdx.x * blockDim.x + threadIdx.x;  // 64*3*64*64 = 786432
  const int j = tid & 63;
  const int i = (tid >> 6) & 63;
  const int c = (tid >> 12) % 3;
  const int p = tid / (3 * 4096);
  const unsigned int idx = sel[p];
  const int ih = (int)(idx / NL);
  const int jwx = (int)(idx % NL);
  const int sh = ih * ST;
  const int sw = jwx * ST;
  const float v = img[(size_t)c * HH * WW + (size_t)(sh + i) * WW + (sw + j)];
  const int orow = (p >> 3) * TT + i;
  const int ocol = (p & 7) * TT + j;
  out[(size_t)c * SIMG * SIMG + (size_t)orow * SIMG + ocol] = v;
}

// ---------------------------------------------------------------------------
extern "C" void kernel_launch(void* const* d_in, const int* in_sizes, int n_in,
                              void* d_out, int out_size, void* d_ws, size_t ws_size,
                              hipStream_t stream) {
  (void)in_sizes; (void)n_in; (void)out_size; (void)ws_size;
  const float* img = (const float*)d_in[0];
  float* out = (float*)d_out;

  // Workspace carve-up (~18.5 MB total)
  float* rs1 = (float*)d_ws;                                    // 4032*512 f32
  float* rs2 = rs1 + (size_t)NROWS_PAD * RS_PITCH;              // 4032*512 f32
  unsigned long long* keys =
      (unsigned long long*)(rs2 + (size_t)NROWS_PAD * RS_PITCH);  // 243049 u64 (8B aligned)
  unsigned int* sel = (unsigned int*)(keys + NTILE);              // 64 u32

  k_rowsum<<<NROWS_PAD, 256, 0, stream>>>(img, rs1, rs2);
  dim3 grid2((NL + 15) / 16, (NL + 15) / 16);                   // 31 x 31 tiles
  k_std_wmma<<<grid2, 32, 0, stream>>>(rs1, rs2, keys);
  k_select<<<1, 1024, 0, stream>>>(keys, sel);
  k_gather<<<(PSEL * 3 * TT * TT) / 256, 256, 0, stream>>>(img, sel, out);
}